// Splatter_61134564491789
// MI455X (gfx1250) — compile-verified
//
#include <hip/hip_runtime.h>
#include <hip/hip_bf16.h>

// Splat 5x5 convolution (full padding): in 8192x8192 f32 -> out 8196x8196 f32.
// Banded-GEMM on the fp32 matrix pipe (V_WMMA_F32_16X16X4_F32):
//   each wave computes a 16x16 output tile, C += A(16x4) x W(4x16),
//   K = 20-wide horizontal input band (5 chunks of 4) x 5 vertical taps
//   => 25 WMMAs per tile, full fp32 (matches reference numerics).
// Input halo staged in LDS (36x68 per 256-thread block -> 32x64 outputs).
// ONE merged LDS allocation: reversed zero-padded filter table at element 0
// (so all B loads are 8-bit-immediate ds_load_2addr_b32 off one register),
// halo tile at element FT_SIZE (A-pair offsets still fit 2addr_b64 imms).

typedef __attribute__((ext_vector_type(2))) float v2f;
typedef __attribute__((ext_vector_type(8))) float v8f;

#define IN_H 8192
#define IN_W 8192
#define OUT_H 8196
#define OUT_W 8196

#define BLK_OY 32              // output rows per block (2 wave-rows)
#define BLK_OX 64              // output cols per block (4 wave-cols)
#define LDS_ROWS 36            // BLK_OY + 4 halo
#define LDS_USED 68            // BLK_OX + 4 halo
#define LDS_COLS 72            // padded stride

#define FT_STRIDE 40           // padded filter-table row (5 rows)
#define FT_SIZE (5 * FT_STRIDE)   // 200 elements, table lives at LDS offset 0
// Reversed band table: gtab[row][q] = f[row][20-q] for q in [16,20], else 0.
// Fragment component v (K index kb = 4kc + 2*half + v) reads q = 16 + kb - n.

__global__ __launch_bounds__(256)
void splat_conv5x5_wmma(const float* __restrict__ in,
                        const float* __restrict__ flt,
                        float* __restrict__ out)
{
    // Single allocation => we control the layout: [0,200) = filter table,
    // [200, 200+36*72) = input halo tile.
    __shared__ __align__(16) float lds[FT_SIZE + LDS_ROWS * LDS_COLS];

    const int tid = threadIdx.x;
    const int by0 = blockIdx.y * BLK_OY;
    const int bx0 = blockIdx.x * BLK_OX;

    // ---- reversed zero-padded banded-filter table (one iteration, tid < 200)
    for (int i = tid; i < FT_SIZE; i += 256) {
        const int row = i / FT_STRIDE;
        const int q   = i - row * FT_STRIDE;
        float v = 0.0f;
        if (q >= 16 && q <= 20)
            v = flt[row * 5 + (20 - q)];
        lds[i] = v;
    }

    // ---- stage zero-padded input halo (rows by0-4..by0+31, cols bx0-4..bx0+63)
    for (int i = tid; i < LDS_ROWS * LDS_COLS; i += 256) {
        const int ly = i / LDS_COLS;
        const int lx = i - ly * LDS_COLS;
        float v = 0.0f;
        if (lx < LDS_USED) {
            const int gy = by0 - 4 + ly;
            const int gx = bx0 - 4 + lx;
            if (gy >= 0 && gy < IN_H && gx >= 0 && gx < IN_W)
                v = in[gy * IN_W + gx];
        }
        lds[FT_SIZE + i] = v;
    }

    __syncthreads();

    // ---- wave / lane geometry (wave32)
    const int lane = tid & 31;
    const int wv   = tid >> 5;
    const int wr   = wv >> 2;          // wave row 0..1
    const int wc   = wv & 3;           // wave col 0..3
    const int half = lane >> 4;        // 0: lanes 0-15, 1: lanes 16-31
    const int n    = lane & 15;        // N col index; also M row index for A

    const int ty0 = by0 + wr * 16;
    const int tx0 = bx0 + wc * 16;
    const int cbase = wc * 16;         // LDS col of band index kk=0

    // Single A base at the MINIMUM row (di=4); all (di,kc) offsets are
    // positive compile-time constants FT_SIZE + (4-di)*LDS_COLS + 4kc (+1),
    // even => mergeable into immediate-offset ds_load_2addr_b64.
    const int abase = (wr * 16 + n) * LDS_COLS + cbase + 2 * half;
    // Single B base at LDS element 0; offsets di*FT_STRIDE + 4kc (+1) <= 197.
    const int bbase = 16 + 2 * half - n;   // in [1,18]

    v8f acc = {};

    #pragma unroll
    for (int di = 0; di < 5; ++di) {
        const int aofs = FT_SIZE + (4 - di) * LDS_COLS;  // constant per di
        const int bofs = di * FT_STRIDE;                 // constant per di

        #pragma unroll
        for (int kc = 0; kc < 5; ++kc) {
            // A fragment: lanes 0-15 hold K = 4kc+{0,1}; lanes 16-31 K = 4kc+{2,3}
            v2f a;
            a.x = lds[abase + aofs + kc * 4];
            a.y = lds[abase + aofs + kc * 4 + 1];

            // B fragment: ascending adjacent reads from the reversed table
            v2f b;
            b.x = lds[bbase + bofs + kc * 4];
            b.y = lds[bbase + bofs + kc * 4 + 1];

            acc = __builtin_amdgcn_wmma_f32_16x16x4_f32(
                /*neg_a=*/false, a, /*neg_b=*/false, b,
                /*c_mod=*/(short)0, acc, /*reuse_a=*/false, /*reuse_b=*/false);
        }
    }

    // ---- store C: VGPR r -> output row (ty0 + 8*half + r), col (tx0 + n)
    #pragma unroll
    for (int r = 0; r < 8; ++r) {
        const int oy = ty0 + half * 8 + r;
        const int ox = tx0 + n;
        if (oy < OUT_H && ox < OUT_W)
            out[oy * OUT_W + ox] = acc[r];
    }
}

extern "C" void kernel_launch(void* const* d_in, const int* in_sizes, int n_in,
                              void* d_out, int out_size, void* d_ws, size_t ws_size,
                              hipStream_t stream) {
    const float* in  = (const float*)d_in[0];   // 8192*8192 f32
    const float* flt = (const float*)d_in[1];   // 25 f32
    // d_in[2] = bias: unused in forward per reference.
    float* out = (float*)d_out;                 // 8196*8196 f32

    dim3 block(256);
    dim3 grid((OUT_W + BLK_OX - 1) / BLK_OX,    // 129
              (OUT_H + BLK_OY - 1) / BLK_OY);   // 257
    splat_conv5x5_wmma<<<grid, block, 0, stream>>>(in, flt, out);
}